// GCN_35424890257988
// MI455X (gfx1250) — compile-verified
//
#include <hip/hip_runtime.h>

typedef __attribute__((ext_vector_type(2))) float v2f;
typedef __attribute__((ext_vector_type(8))) float v8f;

#define D_FEAT 256
#define N_CH   256
#define ROW_TILES 5                       // row-tiles per wave (B-fragment reuse factor)
#define BLK_ROWS (ROW_TILES * 16)         // 80 rows per block; 100000 = 80 * 1250 exactly
#define LDS_STRIDE 260                    // 256 + 4 pad -> conflict-free A-frag ds reads

// ---------------------------------------------------------------------------
// Kernel A: h = X @ W via V_WMMA_F32_16X16X4_F32, out = h*skip + bias.
// Block = 512 threads = 16 waves. Wave w owns N-tile w and FIVE 16-row M-tiles,
// so each B fragment (2 W values/lane) feeds 5 WMMAs -> W L2 traffic /5.
// A-tile (80x256 fp32, 83.2KB padded) staged in LDS, shared by all 16 waves.
// ---------------------------------------------------------------------------
__global__ __launch_bounds__(512)
void gcn_gemm_wmma(const float* __restrict__ feat,
                   const float* __restrict__ kern,
                   const float* __restrict__ bias,
                   const float* __restrict__ skip,
                   float* __restrict__ h,
                   float* __restrict__ out)
{
    __shared__ float alds[BLK_ROWS * LDS_STRIDE];
    const int tid     = threadIdx.x;
    const int rowbase = blockIdx.x * BLK_ROWS;

    // Stage 80x256 feature tile: 20480 floats / 512 threads = 10 x float4 each,
    // fully coalesced (each pass covers 2048 consecutive floats = 8 rows).
#pragma unroll
    for (int i = 0; i < 10; ++i) {
        const int f = i * 2048 + tid * 4;
        const int r = f >> 8;       // row in tile (0..79)
        const int c = f & 255;      // col
        float4 x = *(const float4*)(feat + (size_t)(rowbase + r) * D_FEAT + c);
        *(float4*)&alds[r * LDS_STRIDE + c] = x;
    }
    __syncthreads();

    const int lane  = tid & 31;
    const int ntile = tid >> 5;              // wave id -> N tile 0..15
    const int m     = lane & 15;             // A row-in-tile / C col index
    const int kb    = (lane >> 4) << 1;      // 0 or 2 (K sub-offset per half-wave)
    const int ncol  = ntile * 16 + m;        // global output column

    v8f acc[ROW_TILES];
#pragma unroll
    for (int t = 0; t < ROW_TILES; ++t)
        acc[t] = (v8f){0.f, 0.f, 0.f, 0.f, 0.f, 0.f, 0.f, 0.f};

    const float* arow = &alds[m * LDS_STRIDE + kb];       // + t*16*LDS_STRIDE per tile
    const float* bcol = kern + (size_t)kb * N_CH + ncol;  // W is K-major [256x256]

#pragma unroll 2
    for (int k = 0; k < D_FEAT; k += 4) {
        v2f b;
        b.x = bcol[(size_t)k * N_CH];                     // W[k+kb  ][ncol]
        b.y = bcol[(size_t)k * N_CH + N_CH];              // W[k+kb+1][ncol]
#pragma unroll
        for (int t = 0; t < ROW_TILES; ++t) {
            v2f a = *(const v2f*)(arow + t * 16 * LDS_STRIDE + k);
            acc[t] = __builtin_amdgcn_wmma_f32_16x16x4_f32(
                         false, a, false, b, (short)0, acc[t], false, false);
        }
    }

    const float sk = skip[ncol];
    const float bs = bias[ncol];
    const int   mhalf = (lane >> 4) << 3;                 // +0 or +8 row half
#pragma unroll
    for (int t = 0; t < ROW_TILES; ++t) {
        const int mbase = rowbase + t * 16 + mhalf;
#pragma unroll
        for (int r = 0; r < 8; ++r) {
            const size_t idx = (size_t)(mbase + r) * N_CH + ncol;
            const float  v   = acc[t][r];
            h[idx]   = v;                   // h kept for edge gather (L2-resident)
            out[idx] = v * sk + bs;         // seed pre-activation output
        }
    }
}

// ---------------------------------------------------------------------------
// Kernel B: COO scatter — one wave per edge, lane handles 8 channels (2x b128
// gather of h[col], 8 f32 atomic adds into out[row]). All traffic L2-resident.
// ---------------------------------------------------------------------------
__global__ __launch_bounds__(256)
void gcn_spmm_scatter(const float* __restrict__ h,
                      const int*   __restrict__ erow,
                      const int*   __restrict__ ecol,
                      const float* __restrict__ eval,
                      float* __restrict__ out,
                      int n_edges)
{
    const int wave = (blockIdx.x * 256 + threadIdx.x) >> 5;
    const int lane = threadIdx.x & 31;
    if (wave >= n_edges) return;

    const int   r = erow[wave];
    const int   c = ecol[wave];
    const float v = eval[wave];

    const float4* hp = (const float4*)(h + (size_t)c * N_CH) + lane * 2;
    float4 a = hp[0];
    float4 b = hp[1];

    float* op = out + (size_t)r * N_CH + lane * 8;
    atomicAdd(op + 0, v * a.x);
    atomicAdd(op + 1, v * a.y);
    atomicAdd(op + 2, v * a.z);
    atomicAdd(op + 3, v * a.w);
    atomicAdd(op + 4, v * b.x);
    atomicAdd(op + 5, v * b.y);
    atomicAdd(op + 6, v * b.z);
    atomicAdd(op + 7, v * b.w);
}

// ---------------------------------------------------------------------------
// Kernel C: in-place SELU over out (float4 vectorized).
// ---------------------------------------------------------------------------
__global__ __launch_bounds__(256)
void gcn_selu(float* __restrict__ out, int n4)
{
    const int i = blockIdx.x * 256 + threadIdx.x;
    if (i >= n4) return;
    const float scale = 1.0507009873554805f;
    const float alpha = 1.6732632423543772f;
    float4* p = (float4*)out + i;
    float4 x = *p;
    x.x = x.x > 0.f ? scale * x.x : scale * alpha * (expf(x.x) - 1.f);
    x.y = x.y > 0.f ? scale * x.y : scale * alpha * (expf(x.y) - 1.f);
    x.z = x.z > 0.f ? scale * x.z : scale * alpha * (expf(x.z) - 1.f);
    x.w = x.w > 0.f ? scale * x.w : scale * alpha * (expf(x.w) - 1.f);
    *p = x;
}

// ---------------------------------------------------------------------------
// Launcher: inputs per setup_inputs() order.
// ---------------------------------------------------------------------------
extern "C" void kernel_launch(void* const* d_in, const int* in_sizes, int n_in,
                              void* d_out, int out_size, void* d_ws, size_t ws_size,
                              hipStream_t stream) {
    const float* feat = (const float*)d_in[0];   // [N, 256]
    const int*   erow = (const int*)  d_in[1];   // [E]
    const int*   ecol = (const int*)  d_in[2];   // [E]
    const float* eval = (const float*)d_in[3];   // [E]
    const float* kern = (const float*)d_in[4];   // [256, 256]
    const float* bias = (const float*)d_in[5];   // [256]
    const float* skip = (const float*)d_in[6];   // [256]
    float*       out  = (float*)d_out;           // [N, 256]
    float*       h    = (float*)d_ws;            // [N, 256] scratch (102.4 MB)

    const int n_nodes = in_sizes[0] / D_FEAT;    // 100000 (multiple of 80)
    const int n_edges = in_sizes[1];             // 3200000

    // 1) h = X @ W (WMMA, 5x B-reuse), out = h*skip + bias
    gcn_gemm_wmma<<<n_nodes / BLK_ROWS, 512, 0, stream>>>(feat, kern, bias, skip, h, out);

    // 2) out[r] += val * h[c] per edge (one wave per edge, 8 waves/block)
    const int spmm_blocks = (n_edges + 7) / 8;
    gcn_spmm_scatter<<<spmm_blocks, 256, 0, stream>>>(h, erow, ecol, eval, out, n_edges);

    // 3) out = selu(out)
    const int n4 = n_nodes * (N_CH / 4);
    gcn_selu<<<(n4 + 255) / 256, 256, 0, stream>>>(out, n4);
}